// AttentiveFP_net_678604832933
// MI455X (gfx1250) — compile-verified
//
#include <hip/hip_runtime.h>
#include <hip/hip_bf16.h>
#include <math.h>

#define THREADS 256
static inline int cdiv(long long a, long long b) { return (int)((a + b - 1) / b); }

typedef __attribute__((ext_vector_type(2))) float v2f;
typedef __attribute__((ext_vector_type(8))) float v8f;

#if defined(__HIP_DEVICE_COMPILE__) && defined(__gfx1250__) && __has_builtin(__builtin_amdgcn_wmma_f32_16x16x4_f32)
#define HAVE_WMMA_F32X4 1
#else
#define HAVE_WMMA_F32X4 0
#endif

__device__ __forceinline__ float lrelu(float v) { return v > 0.f ? v : 0.01f * v; }
__device__ __forceinline__ float eluf(float v)  { return v > 0.f ? v : expm1f(v); }
__device__ __forceinline__ float sigf(float v)  { return 1.f / (1.f + expf(-v)); }
// monotone float <-> uint encoding for atomicMax over signed floats
__device__ __forceinline__ unsigned fenc(float f) {
    unsigned u = __float_as_uint(f);
    return (u & 0x80000000u) ? ~u : (u | 0x80000000u);
}
__device__ __forceinline__ float fdec(unsigned u) {
    return (u & 0x80000000u) ? __uint_as_float(u & 0x7fffffffu) : __uint_as_float(~u);
}
#define ENC_NEG_INF 0x007FFFFFu  // fenc(-inf)

// -------------------------------------------------------------------------
// Generic fill
__global__ void k_fill_u32(unsigned* __restrict__ p, unsigned v, int n) {
    int t = blockIdx.x * blockDim.x + threadIdx.x;
    if (t < n) p[t] = v;
}

// -------------------------------------------------------------------------
// xh = leaky(x @ lin1_w.T + lin1_b)   x:[N,25], w:[64,25]
__global__ void k_lin1(const float* __restrict__ x, const float* __restrict__ w,
                       const float* __restrict__ b, float* __restrict__ xh, int N) {
    int t = blockIdx.x * blockDim.x + threadIdx.x;
    if (t >= N * 64) return;
    int i = t >> 6, j = t & 63;
    float acc = b[j];
#pragma unroll
    for (int k = 0; k < 25; ++k) acc += x[i * 25 + k] * w[j * 25 + k];
    xh[t] = lrelu(acc);
}

// -------------------------------------------------------------------------
// Y[M,Nout] = X[M,64] @ W[Nout,64]^T  via V_WMMA_F32_16X16X4_F32.
// One wave owns a 16-row strip: the A fragment (16 x v2f per lane, ISA
// A-layout) stays resident and is reused across all Nout/16 column tiles.
// Column tiles are processed two at a time with explicit B-fragment register
// arrays so the 16 b64 loads per tile issue as a clause and the second tile's
// loads are in flight while the first tile's 16 WMMAs run (no per-WMMA
// load-latency stall).  M % 16 == 0 and Nout/16 even at every call site.
__global__ void __launch_bounds__(256)
k_gemm_xwT(const float* __restrict__ X, const float* __restrict__ W,
           float* __restrict__ Y, int M, int Nout) {
    int wid  = (blockIdx.x * blockDim.x + threadIdx.x) >> 5;
    int lane = threadIdx.x & 31;
    int strips = M >> 4;
    if (wid >= strips) return;              // uniform per wave -> EXEC all-1 inside
    int half = lane >> 4;                   // 0: K pair {0,1}, 1: K pair {2,3}
    int l16  = lane & 15;
    long long rowA = (long long)(wid * 16 + l16) * 64;  // A-matrix row (this lane)
    int rbase = wid * 16 + half * 8;                    // C/D rows owned by this half

    // Preload A strip fragment: 16 K-steps of 4, this lane holds its K pair.
    v2f afrag[16];
#pragma unroll
    for (int s = 0; s < 16; ++s) {
        int ka = s * 4 + half * 2;
        afrag[s].x = X[rowA + ka];
        afrag[s].y = X[rowA + ka + 1];
    }

    int tilesN = Nout >> 4;
    for (int tn = 0; tn < tilesN; tn += 2) {
        long long colB0 = (long long)(tn * 16 + l16) * 64;        // W row, tile tn
        long long colB1 = (long long)((tn + 1) * 16 + l16) * 64;  // W row, tile tn+1
#if HAVE_WMMA_F32X4
        v2f b0[16], b1[16];
#pragma unroll
        for (int s = 0; s < 16; ++s) {
            int ka = s * 4 + half * 2;
            b0[s].x = W[colB0 + ka];
            b0[s].y = W[colB0 + ka + 1];
        }
#pragma unroll
        for (int s = 0; s < 16; ++s) {
            int ka = s * 4 + half * 2;
            b1[s].x = W[colB1 + ka];
            b1[s].y = W[colB1 + ka + 1];
        }
        v8f acc0 = {}, acc1 = {};
#pragma unroll
        for (int s = 0; s < 16; ++s)
            acc0 = __builtin_amdgcn_wmma_f32_16x16x4_f32(false, afrag[s], false, b0[s],
                                                         (short)0, acc0, false, false);
#pragma unroll
        for (int r = 0; r < 8; ++r)
            Y[(long long)(rbase + r) * Nout + tn * 16 + l16] = acc0[r];
#pragma unroll
        for (int s = 0; s < 16; ++s)
            acc1 = __builtin_amdgcn_wmma_f32_16x16x4_f32(false, afrag[s], false, b1[s],
                                                         (short)0, acc1, false, false);
#pragma unroll
        for (int r = 0; r < 8; ++r)
            Y[(long long)(rbase + r) * Nout + (tn + 1) * 16 + l16] = acc1[r];
#else
        float accs0[8] = {0.f, 0.f, 0.f, 0.f, 0.f, 0.f, 0.f, 0.f};
        float accs1[8] = {0.f, 0.f, 0.f, 0.f, 0.f, 0.f, 0.f, 0.f};
        for (int k = 0; k < 64; ++k) {
            float w0 = W[colB0 + k];
            float w1 = W[colB1 + k];
#pragma unroll
            for (int r = 0; r < 8; ++r) {
                float xv = X[(long long)(rbase + r) * 64 + k];
                accs0[r] += xv * w0;
                accs1[r] += xv * w1;
            }
        }
#pragma unroll
        for (int r = 0; r < 8; ++r) {
            Y[(long long)(rbase + r) * Nout + tn * 16 + l16]       = accs0[r];
            Y[(long long)(rbase + r) * Nout + (tn + 1) * 16 + l16] = accs1[r];
        }
#endif
    }
}

// -------------------------------------------------------------------------
// per-row dot with one or two 64-vectors
__global__ void k_node_dot2(const float* __restrict__ feat,
                            const float* __restrict__ a1, float* __restrict__ o1,
                            const float* __restrict__ a2, float* __restrict__ o2, int M) {
    int i = blockIdx.x * blockDim.x + threadIdx.x;
    if (i >= M) return;
    float s1 = 0.f, s2 = 0.f;
#pragma unroll
    for (int k = 0; k < 64; ++k) {
        float v = feat[(long long)i * 64 + k];
        s1 += v * a1[k];
        if (a2) s2 += v * a2[k];
    }
    o1[i] = s1;
    if (o2) o2[i] = s2;
}

// -------------------------------------------------------------------------
// logit = leaky(sa[row] + sb[seg]); record + segment max.  rowidx==null -> row=e
__global__ void k_logit_scatter(const float* __restrict__ sa, const float* __restrict__ sb,
                                const int* __restrict__ rowidx, const int* __restrict__ seg,
                                float* __restrict__ elog, unsigned* __restrict__ segmax,
                                int count) {
    int e = blockIdx.x * blockDim.x + threadIdx.x;
    if (e >= count) return;
    int r = rowidx ? rowidx[e] : e;
    int g = seg[e];
    float a = lrelu(sa[r] + sb[g]);
    elog[e] = a;
    atomicMax(&segmax[g], fenc(a));
}

// exp(logit - segmax) + segment sum
__global__ void k_edge_exp(const float* __restrict__ elog, const int* __restrict__ seg,
                           const unsigned* __restrict__ segmax, float* __restrict__ segsum,
                           float* __restrict__ eexp, int count) {
    int e = blockIdx.x * blockDim.x + threadIdx.x;
    if (e >= count) return;
    int g = seg[e];
    float m = fdec(segmax[g]);
    if (!isfinite(m)) m = 0.f;
    float ex = expf(elog[e] - m);
    eexp[e] = ex;
    atomicAdd(&segsum[g], ex);
}

// out[seg] += feat[row] * (eexp/(segsum+1e-16))  (alpha==1 when eexp==null)
__global__ void k_seg_scatter(const float* __restrict__ feat, const int* __restrict__ rowidx,
                              const int* __restrict__ seg, const float* __restrict__ eexp,
                              const float* __restrict__ segsum, float* __restrict__ out,
                              int count) {
    long long t = (long long)blockIdx.x * blockDim.x + threadIdx.x;
    if (t >= (long long)count * 64) return;
    int e = (int)(t >> 6), j = (int)(t & 63);
    int r = rowidx ? rowidx[e] : e;
    int g = seg[e];
    float w = eexp ? (eexp[e] / (segsum[g] + 1e-16f)) : 1.f;
    atomicAdd(&out[(long long)g * 64 + j], feat[(long long)r * 64 + j] * w);
}

// -------------------------------------------------------------------------
// GATEConv pass 1: m = leaky(cat(xh[src],ea) @ W1^T); logit = leaky(m.att_l + tr[dst])
__global__ void k_gate_logit(const float* __restrict__ xh, const float* __restrict__ ea,
                             const int* __restrict__ src, const int* __restrict__ dst,
                             const float* __restrict__ W1, const float* __restrict__ attl,
                             const float* __restrict__ tr, float* __restrict__ elog,
                             unsigned* __restrict__ nmax, int E) {
    int e = blockIdx.x * blockDim.x + threadIdx.x;
    if (e >= E) return;
    int s = src[e], d = dst[e];
    float cat[68];
#pragma unroll
    for (int k = 0; k < 64; ++k) cat[k] = xh[(long long)s * 64 + k];
#pragma unroll
    for (int k = 0; k < 4; ++k) cat[64 + k] = ea[(long long)e * 4 + k];
    float lg = 0.f;
    for (int j = 0; j < 64; ++j) {
        float acc = 0.f;
#pragma unroll
        for (int k = 0; k < 68; ++k) acc += cat[k] * W1[j * 68 + k];
        lg += attl[j] * lrelu(acc);
    }
    float a = lrelu(lg + tr[d]);
    elog[e] = a;
    atomicMax(&nmax[d], fenc(a));
}

// GATEConv pass 3: pre[dst] += alpha * m   (agg = pre @ W2^T done by WMMA GEMM after)
__global__ void k_gate_scatter(const float* __restrict__ xh, const float* __restrict__ ea,
                               const int* __restrict__ src, const int* __restrict__ dst,
                               const float* __restrict__ W1, const float* __restrict__ eexp,
                               const float* __restrict__ nsum, float* __restrict__ pre, int E) {
    int e = blockIdx.x * blockDim.x + threadIdx.x;
    if (e >= E) return;
    int s = src[e], d = dst[e];
    float alpha = eexp[e] / (nsum[d] + 1e-16f);
    float cat[68];
#pragma unroll
    for (int k = 0; k < 64; ++k) cat[k] = xh[(long long)s * 64 + k];
#pragma unroll
    for (int k = 0; k < 4; ++k) cat[64 + k] = ea[(long long)e * 4 + k];
    for (int j = 0; j < 64; ++j) {
        float acc = 0.f;
#pragma unroll
        for (int k = 0; k < 68; ++k) acc += cat[k] * W1[j * 68 + k];
        atomicAdd(&pre[(long long)d * 64 + j], alpha * lrelu(acc));
    }
}

// -------------------------------------------------------------------------
// p = elu(p + bias[j])
__global__ void k_elu_bias(float* __restrict__ p, const float* __restrict__ bias, int M) {
    int t = blockIdx.x * blockDim.x + threadIdx.x;
    if (t >= M * 64) return;
    p[t] = eluf(p[t] + bias[t & 63]);
}

__global__ void k_relu(float* __restrict__ p, int n) {
    int t = blockIdx.x * blockDim.x + threadIdx.x;
    if (t < n) p[t] = p[t] > 0.f ? p[t] : 0.f;
}

// GRUCell + relu:  gi = xin@wih^T (no bias yet), gh = h@whh^T (no bias yet)
__global__ void k_gru_relu(const float* __restrict__ gi, const float* __restrict__ gh,
                           const float* __restrict__ bih, const float* __restrict__ bhh,
                           const float* __restrict__ hprev, float* __restrict__ out, int M) {
    int t = blockIdx.x * blockDim.x + threadIdx.x;
    if (t >= M * 64) return;
    int i = t >> 6, j = t & 63;
    long long base = (long long)i * 192;
    float ir = gi[base + j]        + bih[j];
    float iz = gi[base + 64 + j]   + bih[64 + j];
    float in = gi[base + 128 + j]  + bih[128 + j];
    float hr = gh[base + j]        + bhh[j];
    float hz = gh[base + 64 + j]   + bhh[64 + j];
    float hn = gh[base + 128 + j]  + bhh[128 + j];
    float r = sigf(ir + hr);
    float z = sigf(iz + hz);
    float n = tanhf(in + r * hn);
    float v = (1.f - z) * n + z * hprev[t];
    out[t] = v > 0.f ? v : 0.f;
}

// res[g] = out[g] . lin2_w + lin2_b
__global__ void k_final(const float* __restrict__ out, const float* __restrict__ w,
                        const float* __restrict__ b, float* __restrict__ res, int G) {
    int g = blockIdx.x * blockDim.x + threadIdx.x;
    if (g >= G) return;
    float acc = b[0];
#pragma unroll
    for (int k = 0; k < 64; ++k) acc += out[g * 64 + k] * w[k];
    res[g] = acc;
}

// =========================================================================
extern "C" void kernel_launch(void* const* d_in, const int* in_sizes, int n_in,
                              void* d_out, int out_size, void* d_ws, size_t ws_size,
                              hipStream_t stream) {
    const float* x        = (const float*)d_in[0];
    const int*   ei       = (const int*)d_in[1];
    const float* ea       = (const float*)d_in[2];
    const int*   batch    = (const int*)d_in[3];
    const float* lin1_w   = (const float*)d_in[4];
    const float* lin1_b   = (const float*)d_in[5];
    const float* g_attl   = (const float*)d_in[6];
    const float* g_attr   = (const float*)d_in[7];
    const float* g_lin1w  = (const float*)d_in[8];
    const float* g_lin2w  = (const float*)d_in[9];
    const float* g_bias   = (const float*)d_in[10];
    const float* gru0_wih = (const float*)d_in[11];
    const float* gru0_whh = (const float*)d_in[12];
    const float* gru0_bih = (const float*)d_in[13];
    const float* gru0_bhh = (const float*)d_in[14];
    const float* a_linw   = (const float*)d_in[15];  // [4,64,64]
    const float* a_attsrc = (const float*)d_in[16];  // [4,64]
    const float* a_attdst = (const float*)d_in[17];
    const float* a_bias   = (const float*)d_in[18];
    const float* a_wih    = (const float*)d_in[19];  // [4,192,64]
    const float* a_whh    = (const float*)d_in[20];
    const float* a_bih    = (const float*)d_in[21];  // [4,192]
    const float* a_bhh    = (const float*)d_in[22];
    const float* m_linw   = (const float*)d_in[23];
    const float* m_attsrc = (const float*)d_in[24];
    const float* m_attdst = (const float*)d_in[25];
    const float* m_bias   = (const float*)d_in[26];
    const float* m_wih    = (const float*)d_in[27];
    const float* m_whh    = (const float*)d_in[28];
    const float* m_bih    = (const float*)d_in[29];
    const float* m_bhh    = (const float*)d_in[30];
    const float* lin2_w   = (const float*)d_in[31];
    const float* lin2_b   = (const float*)d_in[32];

    const int N = in_sizes[3];
    const int E = in_sizes[1] / 2;
    const int G = out_size;
    const int* src = ei;
    const int* dst = ei + E;
    const int L = in_sizes[16] / 64;   // 4 atom layers

    // ---- workspace partition (floats) ----
    float* F = (float*)d_ws;
    size_t o = 0;
    auto alloc = [&](size_t n) { float* p = F + o; o += n; return p; };
    float* xh   = alloc((size_t)N * 64);
    float* xcA  = alloc((size_t)N * 64);
    float* xcB  = alloc((size_t)N * 64);
    float* xt   = alloc((size_t)N * 64);    // also reused as xs in readout
    float* pre  = alloc((size_t)N * 64);
    float* agg  = alloc((size_t)N * 64);
    float* gi   = alloc((size_t)N * 192);
    float* gh   = alloc((size_t)N * 192);
    float* ssrc = alloc(N);
    float* sdst = alloc(N);
    unsigned* nmax = (unsigned*)alloc(N);
    float* nsum = alloc(N);
    float* elog = alloc(E);                 // also reused (size N) in readout
    float* eexp = alloc(E);
    float* asrc = alloc(N);
    float* outA = alloc((size_t)G * 64);
    float* outB = alloc((size_t)G * 64);
    float* xd   = alloc((size_t)G * 64);
    float* tdst = alloc(G);
    unsigned* gmax = (unsigned*)alloc(G);
    float* gsum = alloc(G);
    float* aggG = alloc((size_t)G * 64);
    float* giG  = alloc((size_t)G * 192);
    float* ghG  = alloc((size_t)G * 192);
    (void)ws_size; (void)n_in;

    auto gemm = [&](const float* X, const float* W, float* Y, int M, int Nout) {
        int waves = M / 16;   // one wave per 16-row strip, loops over Nout tiles
        k_gemm_xwT<<<cdiv((long long)waves * 32, THREADS), THREADS, 0, stream>>>(X, W, Y, M, Nout);
    };
    auto fillF = [&](float* p, float v, long long n) {
        k_fill_u32<<<cdiv(n, THREADS), THREADS, 0, stream>>>((unsigned*)p, __builtin_bit_cast(unsigned, v), (int)n);
    };
    auto fillU = [&](unsigned* p, unsigned v, long long n) {
        k_fill_u32<<<cdiv(n, THREADS), THREADS, 0, stream>>>(p, v, (int)n);
    };

    // ---- 1) xh = leaky(x @ lin1_w^T + b) ----
    k_lin1<<<cdiv((long long)N * 64, THREADS), THREADS, 0, stream>>>(x, lin1_w, lin1_b, xh, N);

    // ---- 2) GATEConv + GRU0 ----
    k_node_dot2<<<cdiv(N, THREADS), THREADS, 0, stream>>>(xh, g_attr, sdst, nullptr, nullptr, N);
    fillU(nmax, ENC_NEG_INF, N);
    fillF(nsum, 0.f, N);
    fillF(pre, 0.f, (long long)N * 64);
    k_gate_logit<<<cdiv(E, THREADS), THREADS, 0, stream>>>(xh, ea, src, dst, g_lin1w, g_attl, sdst, elog, nmax, E);
    k_edge_exp<<<cdiv(E, THREADS), THREADS, 0, stream>>>(elog, dst, nmax, nsum, eexp, E);
    k_gate_scatter<<<cdiv(E, THREADS), THREADS, 0, stream>>>(xh, ea, src, dst, g_lin1w, eexp, nsum, pre, E);
    gemm(pre, g_lin2w, agg, N, 64);                        // agg = pre @ W2^T  (WMMA)
    k_elu_bias<<<cdiv((long long)N * 64, THREADS), THREADS, 0, stream>>>(agg, g_bias, N);
    gemm(agg, gru0_wih, gi, N, 192);                       // (WMMA)
    gemm(xh,  gru0_whh, gh, N, 192);                       // (WMMA)
    k_gru_relu<<<cdiv((long long)N * 64, THREADS), THREADS, 0, stream>>>(gi, gh, gru0_bih, gru0_bhh, xh, xcA, N);

    // ---- 3) atom GAT + GRU layers ----
    float* xc_in = xcA;
    float* xc_out = xcB;
    for (int l = 0; l < L; ++l) {
        gemm(xc_in, a_linw + (size_t)l * 64 * 64, xt, N, 64);   // xt = xc @ Wl^T (WMMA)
        k_node_dot2<<<cdiv(N, THREADS), THREADS, 0, stream>>>(xt, a_attsrc + l * 64, ssrc,
                                                             a_attdst + l * 64, sdst, N);
        fillU(nmax, ENC_NEG_INF, N);
        fillF(nsum, 0.f, N);
        fillF(agg, 0.f, (long long)N * 64);
        k_logit_scatter<<<cdiv(E, THREADS), THREADS, 0, stream>>>(ssrc, sdst, src, dst, elog, nmax, E);
        k_edge_exp<<<cdiv(E, THREADS), THREADS, 0, stream>>>(elog, dst, nmax, nsum, eexp, E);
        k_seg_scatter<<<cdiv((long long)E * 64, THREADS), THREADS, 0, stream>>>(xt, src, dst, eexp, nsum, agg, E);
        k_elu_bias<<<cdiv((long long)N * 64, THREADS), THREADS, 0, stream>>>(agg, a_bias + l * 64, N);
        gemm(agg,   a_wih + (size_t)l * 192 * 64, gi, N, 192);  // (WMMA)
        gemm(xc_in, a_whh + (size_t)l * 192 * 64, gh, N, 192);  // (WMMA)
        k_gru_relu<<<cdiv((long long)N * 64, THREADS), THREADS, 0, stream>>>(
            gi, gh, a_bih + l * 192, a_bhh + l * 192, xc_in, xc_out, N);
        float* tmp = xc_in; xc_in = xc_out; xc_out = tmp;
    }

    // ---- 4) attentive readout ----
    fillF(outA, 0.f, (long long)G * 64);
    k_seg_scatter<<<cdiv((long long)N * 64, THREADS), THREADS, 0, stream>>>(
        xc_in, nullptr, batch, nullptr, nullptr, outA, N);
    k_relu<<<cdiv((long long)G * 64, THREADS), THREADS, 0, stream>>>(outA, G * 64);
    gemm(xc_in, m_linw, xt, N, 64);                         // xs = xc @ mol_lin^T (WMMA)
    k_node_dot2<<<cdiv(N, THREADS), THREADS, 0, stream>>>(xt, m_attsrc, asrc, nullptr, nullptr, N);

    float* out_in = outA;
    float* out_out = outB;
    for (int t = 0; t < 3; ++t) {
        gemm(out_in, m_linw, xd, G, 64);                    // (WMMA)
        k_node_dot2<<<cdiv(G, THREADS), THREADS, 0, stream>>>(xd, m_attdst, tdst, nullptr, nullptr, G);
        fillU(gmax, ENC_NEG_INF, G);
        fillF(gsum, 0.f, G);
        fillF(aggG, 0.f, (long long)G * 64);
        k_logit_scatter<<<cdiv(N, THREADS), THREADS, 0, stream>>>(asrc, tdst, nullptr, batch, elog, gmax, N);
        k_edge_exp<<<cdiv(N, THREADS), THREADS, 0, stream>>>(elog, batch, gmax, gsum, eexp, N);
        k_seg_scatter<<<cdiv((long long)N * 64, THREADS), THREADS, 0, stream>>>(
            xt, nullptr, batch, eexp, gsum, aggG, N);
        k_elu_bias<<<cdiv((long long)G * 64, THREADS), THREADS, 0, stream>>>(aggG, m_bias, G);
        gemm(aggG,   m_wih, giG, G, 192);                   // (WMMA)
        gemm(out_in, m_whh, ghG, G, 192);                   // (WMMA)
        k_gru_relu<<<cdiv((long long)G * 64, THREADS), THREADS, 0, stream>>>(
            giG, ghG, m_bih, m_bhh, out_in, out_out, G);
        float* tmp = out_in; out_in = out_out; out_out = tmp;
    }

    // ---- 5) final linear ----
    k_final<<<cdiv(G, THREADS), THREADS, 0, stream>>>(out_in, lin2_w, lin2_b, (float*)d_out, G);
}